// GCNLayer_4423816315102
// MI455X (gfx1250) — compile-verified
//
#include <hip/hip_runtime.h>
#include <hip/hip_bf16.h>

typedef float v2f __attribute__((ext_vector_type(2)));
typedef float v8f __attribute__((ext_vector_type(8)));

#define IN_FEATS 256
#define OUT_FEATS 128

// ---------------------------------------------------------------------------
// Zero a float buffer (grid-stride).
// ---------------------------------------------------------------------------
__global__ __launch_bounds__(256) void zero_kernel(float* __restrict__ p, long long n) {
    long long i = (long long)blockIdx.x * blockDim.x + threadIdx.x;
    long long stride = (long long)gridDim.x * blockDim.x;
    for (; i < n; i += stride) p[i] = 0.0f;
}

// ---------------------------------------------------------------------------
// Degree accumulation: deg_src[src[e]] += 1, deg_dst[dst[e]] += 1.
// fp32 counters are exact up to 2^24 > 3.2M edges.
// ---------------------------------------------------------------------------
__global__ __launch_bounds__(256) void degree_kernel(const int* __restrict__ src,
                                                     const int* __restrict__ dst,
                                                     float* __restrict__ deg_src,
                                                     float* __restrict__ deg_dst,
                                                     int n_edges) {
    int e = blockIdx.x * blockDim.x + threadIdx.x;
    if (e >= n_edges) return;
    atomicAdd(&deg_src[src[e]], 1.0f);
    atomicAdd(&deg_dst[dst[e]], 1.0f);
}

// ---------------------------------------------------------------------------
// In-place deg -> rsqrt(max(deg, 1)).
// ---------------------------------------------------------------------------
__global__ __launch_bounds__(256) void norm_kernel(float* __restrict__ deg, int n) {
    int i = blockIdx.x * blockDim.x + threadIdx.x;
    if (i >= n) return;
    deg[i] = rsqrtf(fmaxf(deg[i], 1.0f));
}

// ---------------------------------------------------------------------------
// h = (x * norm_src[:,None]) @ W   via V_WMMA_F32_16X16X4_F32.
// One wave per 16-row block; 8 column tiles of 16 (OUT_FEATS=128);
// K=256 consumed 4 at a time -> 64 * 8 = 512 WMMA per wave.
//
// Fragment layouts (wave32, ISA 7.12.2):
//   A 16x4:  lane<16 -> M=lane, K={0,1} in {a.x,a.y}; lane>=16 -> K={2,3}
//   B 4x16:  lane<16 -> N=lane, K={0,1} in {b.x,b.y}; lane>=16 -> K={2,3}
//   C/D:     VGPR v: lanes 0-15 M=v, lanes 16-31 M=v+8; N=lane%16
// ---------------------------------------------------------------------------
__global__ __launch_bounds__(256) void gemm_wmma_kernel(const float* __restrict__ x,
                                                        const float* __restrict__ W,
                                                        const float* __restrict__ norm_src,
                                                        float* __restrict__ h,
                                                        int n_nodes) {
    const int lane  = threadIdx.x & 31;
    const int wave  = threadIdx.x >> 5;
    const int rblk  = blockIdx.x * 8 + wave;   // 16-row block index
    const int row0  = rblk * 16;
    if (row0 >= n_nodes) return;

    const int mrow  = lane & 15;               // M (for A) / N (for B) this lane covers
    const int khalf = (lane >> 4) << 1;        // 0 for lanes 0-15, 2 for lanes 16-31

    const float ns = norm_src[row0 + mrow];
    const float* __restrict__ xrow = x + (long long)(row0 + mrow) * IN_FEATS;

    v8f acc[8] = {};

    for (int k = 0; k < IN_FEATS; k += 4) {
        v2f a;
        a.x = xrow[k + khalf + 0] * ns;        // fold D_out^{-1/2} into A
        a.y = xrow[k + khalf + 1] * ns;
#pragma unroll
        for (int t = 0; t < 8; ++t) {
            const int col = t * 16 + mrow;
            v2f b;
            b.x = W[(long long)(k + khalf + 0) * OUT_FEATS + col];
            b.y = W[(long long)(k + khalf + 1) * OUT_FEATS + col];
            acc[t] = __builtin_amdgcn_wmma_f32_16x16x4_f32(
                /*neg_a=*/false, a, /*neg_b=*/false, b,
                /*c_mod=*/(short)0, acc[t], /*reuse_a=*/false, /*reuse_b=*/false);
        }
    }

    const int outRow0 = row0 + ((lane >> 4) << 3);  // +0 (lanes 0-15) or +8 (lanes 16-31)
#pragma unroll
    for (int t = 0; t < 8; ++t) {
#pragma unroll
        for (int v = 0; v < 8; ++v) {
            h[(long long)(outRow0 + v) * OUT_FEATS + t * 16 + mrow] = acc[t][v];
        }
    }
}

// ---------------------------------------------------------------------------
// SpMM scatter: agg[dst[e]][:] += h[src[e]][:].
// One wave per edge; each lane moves 4 consecutive floats (32*4 = 128).
// ---------------------------------------------------------------------------
__global__ __launch_bounds__(256) void scatter_kernel(const float* __restrict__ h,
                                                      const int* __restrict__ src,
                                                      const int* __restrict__ dst,
                                                      float* __restrict__ agg,
                                                      int n_edges) {
    const int lane = threadIdx.x & 31;
    const int edge = (int)(((long long)blockIdx.x * blockDim.x + threadIdx.x) >> 5);
    if (edge >= n_edges) return;
    const int s = src[edge];
    const int d = dst[edge];
    const float4 v = *(const float4*)(h + (long long)s * OUT_FEATS + lane * 4);
    float* out = agg + (long long)d * OUT_FEATS + lane * 4;
    atomicAdd(out + 0, v.x);
    atomicAdd(out + 1, v.y);
    atomicAdd(out + 2, v.z);
    atomicAdd(out + 3, v.w);
}

// ---------------------------------------------------------------------------
// out = relu(agg * norm_dst[row] + b[col]), in place on d_out.
// ---------------------------------------------------------------------------
__global__ __launch_bounds__(256) void finalize_kernel(float* __restrict__ out,
                                                       const float* __restrict__ norm_dst,
                                                       const float* __restrict__ bias,
                                                       long long total) {
    long long i = (long long)blockIdx.x * blockDim.x + threadIdx.x;
    if (i >= total) return;
    const int row = (int)(i >> 7);          // / OUT_FEATS
    const int col = (int)(i & (OUT_FEATS - 1));
    float v = out[i] * norm_dst[row] + bias[col];
    out[i] = fmaxf(v, 0.0f);
}

// ---------------------------------------------------------------------------
// Inputs (setup_inputs order): x, W, b, src, dst, layer_id, n_layers
// ---------------------------------------------------------------------------
extern "C" void kernel_launch(void* const* d_in, const int* in_sizes, int n_in,
                              void* d_out, int out_size, void* d_ws, size_t ws_size,
                              hipStream_t stream) {
    const float* x    = (const float*)d_in[0];
    const float* W    = (const float*)d_in[1];
    const float* bias = (const float*)d_in[2];
    const int*   src  = (const int*)d_in[3];
    const int*   dst  = (const int*)d_in[4];

    const int n_nodes = in_sizes[0] / IN_FEATS;
    const int n_edges = in_sizes[3];

    // Workspace layout (floats): [norm_src | norm_dst | h]
    float* wsf      = (float*)d_ws;
    float* norm_src = wsf;
    float* norm_dst = wsf + n_nodes;
    float* h        = wsf + 2LL * n_nodes;     // 2*100000*4 B = 16B-aligned
    float* agg      = (float*)d_out;           // accumulate in place

    const long long deg_total = 2LL * n_nodes;
    const long long out_total = (long long)n_nodes * OUT_FEATS;

    // 1) zero degree counters and output accumulator
    zero_kernel<<<(int)((deg_total + 255) / 256), 256, 0, stream>>>(norm_src, deg_total);
    zero_kernel<<<(int)((out_total + 255) / 256), 256, 0, stream>>>(agg, out_total);

    // 2) degrees
    degree_kernel<<<(n_edges + 255) / 256, 256, 0, stream>>>(src, dst, norm_src, norm_dst, n_edges);

    // 3) rsqrt(max(deg,1)) in place (both buffers are contiguous)
    norm_kernel<<<(int)((deg_total + 255) / 256), 256, 0, stream>>>(norm_src, (int)deg_total);

    // 4) WMMA GEMM: h = (x * norm_src) @ W   (6250 row-blocks, 8 waves/block)
    const int rblks = (n_nodes + 15) / 16;
    gemm_wmma_kernel<<<(rblks + 7) / 8, 256, 0, stream>>>(x, W, norm_src, h, n_nodes);

    // 5) edge scatter: one wave per edge
    const long long scatter_threads = (long long)n_edges * 32;
    scatter_kernel<<<(int)((scatter_threads + 255) / 256), 256, 0, stream>>>(h, src, dst, agg, n_edges);

    // 6) scale + bias + relu
    finalize_kernel<<<(int)((out_total + 255) / 256), 256, 0, stream>>>(agg, norm_dst, bias, out_total);
}